// QueryAndGroup_28707561406745
// MI455X (gfx1250) — compile-verified
//
#include <hip/hip_runtime.h>

// ---------------------------------------------------------------------------
// QueryAndGroup (pointnet2) for MI455X / gfx1250, wave32.
//   Kernel 1: ball query. Distance matrix via V_WMMA_F32_16X16X4_F32
//             (16 centers x 16 points per instruction). Zero accumulator
//             (inline SRC2=0); |p|^2 folded into the compare threshold.
//             Software-pipelined point-tile loads. In-order first-32
//             selection via wave32 ballots, early exit, fill pass.
//   Kernel 2: grouping gather (coalesced writes, L2-resident gathers).
// ---------------------------------------------------------------------------

typedef __attribute__((ext_vector_type(2))) float v2f;
typedef __attribute__((ext_vector_type(8))) float v8f;

#define NSAMPLE 32

__global__ void __launch_bounds__(256)
ball_query_wmma(const float* __restrict__ xyz,      // (B,N,3)
                const float* __restrict__ new_xyz,  // (B,S,3)
                int* __restrict__ idx_out,          // (B,S,32)
                int B, int N, int S, float r2)
{
    const int lane   = threadIdx.x & 31;
    const int waveId = (blockIdx.x * blockDim.x + threadIdx.x) >> 5;
    const int tilesPerBatch = S >> 4;             // 16 centers per wave
    const int b  = waveId / tilesPerBatch;
    const int ct = waveId % tilesPerBatch;
    if (b >= B) return;
    const int cBase = ct << 4;
    const int m = lane & 15;                       // both halves mirror 0..15

    // ---- A operand: center tile. 16x4 f32 A layout:
    //   lanes 0-15 : VGPR0=K0, VGPR1=K1 ; lanes 16-31 : VGPR0=K2, VGPR1=K3
    // Row m = (-2cx, -2cy, -2cz, |c|^2); B col n = (px,py,pz,1)  =>
    // D[m][n] = |c|^2 - 2 c.p  ;  hit iff D[m][n] < r2 - |p|^2
    const float* cp = new_xyz + ((size_t)b * S + cBase + m) * 3;
    const float cx = cp[0], cy = cp[1], cz = cp[2];
    const float cc = cx * cx + cy * cy + cz * cz;
    v2f A;
    A.x = (lane < 16) ? (-2.0f * cx) : (-2.0f * cz);
    A.y = (lane < 16) ? (-2.0f * cy) : cc;

    int cnt_v   = 0;   // lane l (0..15) holds hit count of row l
    int first_v = 0;   // lane l (0..15) holds first hit index of row l (0 if none)

    const unsigned halfmask = (lane < 16) ? 0x0000FFFFu : 0xFFFF0000u;
    const unsigned ltmask   = ((lane == 31) ? 0x7FFFFFFFu : ((1u << lane) - 1u)) & halfmask;
    const int      rowBaseAdd = (lane < 16) ? 0 : 8;

    const float* pbase = xyz + (size_t)b * N * 3;

    // ---- software pipeline: preload tile 0
    float px, py, pz;
    {
        const float* q = pbase + (size_t)m * 3;
        px = q[0]; py = q[1]; pz = q[2];
    }

    for (int p0 = 0; p0 < N; p0 += 16) {
        // issue next tile's load early (clamped; redundant load if last tile)
        const int pn = (p0 + 16 < N) ? (p0 + 16) : 0;
        const float* q = pbase + (size_t)(pn + m) * 3;
        const float nx = q[0], ny = q[1], nz = q[2];

        // ---- B operand for current tile
        v2f Bop;
        Bop.x = (lane < 16) ? px : pz;
        Bop.y = (lane < 16) ? py : 1.0f;
        const float pp  = px * px + py * py + pz * pz;
        const float thr = r2 - pp;                 // per-column threshold

        // D = A*B + 0  (SRC2 inline zero)
        const v8f Z = {0.f, 0.f, 0.f, 0.f, 0.f, 0.f, 0.f, 0.f};
        v8f D = __builtin_amdgcn_wmma_f32_16x16x4_f32(
            false, A, false, Bop, (short)0, Z, false, false);

        // ---- selection: VGPR r holds rows r (lanes 0-15) and r+8 (lanes 16-31)
#pragma unroll
        for (int r = 0; r < 8; ++r) {
            const unsigned mask = (unsigned)__ballot(D[r] < thr);
            if (mask == 0u) continue;
            const int row     = r + rowBaseAdd;               // per-half row id
            const int basecnt = __shfl(cnt_v, row);           // counter of my row
            const int slot    = basecnt + __popc(mask & ltmask);
            const bool hit    = (mask >> lane) & 1u;
            if (hit && slot < NSAMPLE)
                idx_out[((size_t)b * S + cBase + row) * NSAMPLE + slot] = p0 + m;
            // update per-row counters / first-hit (rows r, r+8 live in lanes r, r+8)
            const int pcLo = __popc(mask & 0x0000FFFFu);
            const int pcHi = __popc(mask >> 16);
            if (lane == r) {
                if (cnt_v == 0 && pcLo) first_v = p0 + (__ffs((int)(mask & 0xFFFFu)) - 1);
                cnt_v += pcLo;
            }
            if (lane == r + 8) {
                if (cnt_v == 0 && pcHi) first_v = p0 + (__ffs((int)(mask >> 16)) - 1);
                cnt_v += pcHi;
            }
        }

        // ---- early exit once all 16 rows have >= NSAMPLE hits
        const unsigned done = (unsigned)__ballot(cnt_v >= NSAMPLE);
        if ((done & 0xFFFFu) == 0xFFFFu) break;

        px = nx; py = ny; pz = nz;                 // rotate pipeline
    }

    // ---- fill pass: slots >= cnt get first index (0 if none found)
    for (int row = 0; row < 16; ++row) {
        const int cm = __shfl(cnt_v, row);
        const int fv = __shfl(first_v, row);
        if (lane >= cm)
            idx_out[((size_t)b * S + cBase + row) * NSAMPLE + lane] = fv;
    }
}

__global__ void __launch_bounds__(256)
group_gather(const float* __restrict__ xyz,       // (B,N,3)
             const float* __restrict__ new_xyz,   // (B,S,3)
             const float* __restrict__ feat,      // (B,C,N)
             const int* __restrict__ idx,         // (B,S,32)
             float* __restrict__ out_xyz,         // (B,3,S,32)
             float* __restrict__ out_feat,        // (B,C,S,32)
             int B, int N, int S, int C)
{
    const int lane = threadIdx.x & 31;                         // sample slot j
    const int wave = (blockIdx.x * blockDim.x + threadIdx.x) >> 5;
    if (wave >= B * S) return;
    const int b = wave / S;
    const int s = wave % S;

    const int id = idx[((size_t)b * S + s) * NSAMPLE + lane];

    const float* p = xyz + ((size_t)b * N + id) * 3;
    const float px = p[0], py = p[1], pz = p[2];
    const float* cp = new_xyz + ((size_t)b * S + s) * 3;       // uniform, L0 hit
    const float cx = cp[0], cy = cp[1], cz = cp[2];

    const size_t SW = (size_t)S * NSAMPLE;
    const size_t ox = (((size_t)b * 3) * S + s) * NSAMPLE + lane;
    out_xyz[ox]          = px - cx;                            // coalesced
    out_xyz[ox + SW]     = py - cy;
    out_xyz[ox + 2 * SW] = pz - cz;

    const float* fb = feat + (size_t)b * C * N;
    const size_t of = (((size_t)b * C) * S + s) * NSAMPLE + lane;
#pragma unroll 4
    for (int c = 0; c < C; ++c)
        out_feat[of + (size_t)c * SW] = fb[(size_t)c * N + id];
}

extern "C" void kernel_launch(void* const* d_in, const int* in_sizes, int n_in,
                              void* d_out, int out_size, void* d_ws, size_t ws_size,
                              hipStream_t stream) {
    const float* xyz     = (const float*)d_in[0];   // (B,N,3)
    const float* new_xyz = (const float*)d_in[1];   // (B,S,3)
    const float* feat    = (const float*)d_in[2];   // (B,C,N)

    const int B = 4;
    const int N = in_sizes[0] / (B * 3);
    const int S = in_sizes[1] / (B * 3);
    const int C = in_sizes[2] / (B * N);
    const float r2 = 0.1f * 0.1f;

    float* out_xyz  = (float*)d_out;
    float* out_feat = out_xyz + (size_t)B * 3 * S * NSAMPLE;
    int*   idx      = (int*)d_ws;                   // B*S*32 ints (2 MB)

    // Kernel 1: one wave per 16 centers
    {
        const int waves   = (B * S) / 16;
        const int threads = waves * 32;
        dim3 blk(256), grd((threads + 255) / 256);
        ball_query_wmma<<<grd, blk, 0, stream>>>(xyz, new_xyz, idx, B, N, S, r2);
    }
    // Kernel 2: one wave per center
    {
        const int waves   = B * S;
        const int threads = waves * 32;
        dim3 blk(256), grd((threads + 255) / 256);
        group_gather<<<grd, blk, 0, stream>>>(xyz, new_xyz, feat, idx,
                                              out_xyz, out_feat, B, N, S, C);
    }
}